// RetentionEncoder_34660386079033
// MI455X (gfx1250) — compile-verified
//
#include <hip/hip_runtime.h>
#include <hip/hip_bf16.h>
#include <math.h>
#include <stdint.h>

#define L_SEQ    768
#define SEQ_D    384
#define PAIR_D   128
#define NH       8
#define HD       32      // head dim (KD == VD)
#define HKD      256     // NH*HD
#define EPSV     1e-5f

typedef float v2f  __attribute__((ext_vector_type(2)));
typedef float v8f  __attribute__((ext_vector_type(8)));

// Device pass only: host pass doesn't declare amdgcn builtins.
#if defined(__HIP_DEVICE_COMPILE__) && !__has_builtin(__builtin_amdgcn_wmma_f32_16x16x4_f32)
#error "expected gfx1250 f32 WMMA builtin (confirmed present in round 1)"
#endif

__device__ __forceinline__ void wait_asynccnt0() {
#if __has_builtin(__builtin_amdgcn_s_wait_asynccnt)
    __builtin_amdgcn_s_wait_asynccnt(0);
#else
    asm volatile("s_wait_asynccnt 0" ::: "memory");
#endif
}

// ---------------------------------------------------------------------------
// Kernel 0: zero the atomic accumulators (m1 raw sums + groupnorm stats)
// ---------------------------------------------------------------------------
__global__ void zero_buf_kernel(float* __restrict__ m1, float* __restrict__ stats) {
    const int idx = blockIdx.x * 256 + threadIdx.x;
    if (idx < L_SEQ * PAIR_D) m1[idx] = 0.f;
    if (idx < 16) stats[idx] = 0.f;
}

// ---------------------------------------------------------------------------
// Kernel 1: single fused pass over pair[768][768][128] (302 MB -> HBM bound)
//   m2[i][c] = mean_j pair[i][j][c]   (register acc + LDS reduce, scaled here)
//   m1[j][c] = sum_i  pair[i][j][c]   (16-i partial in regs, then f32 atomics;
//                                      scaled later in the GEMM epilogue)
// grid: 48 blocks (16 i's each) x 256 threads; float4 loads + GL2 prefetch.
// ---------------------------------------------------------------------------
__global__ void pair_reduce_kernel(const float* __restrict__ pair,
                                   float* __restrict__ m1_sum,
                                   float* __restrict__ m2) {
    const int t  = threadIdx.x;     // 256
    const int c4 = t & 31;          // float4 slot within 128-channel row
    const int jg = t >> 5;          // 0..7 : j-subgroup
    const int i0 = blockIdx.x * 16;

    float4 m2acc[16];
#pragma unroll
    for (int i = 0; i < 16; ++i) m2acc[i] = make_float4(0.f, 0.f, 0.f, 0.f);

    for (int jb = 0; jb < L_SEQ / 8; ++jb) {
        const int j = jb * 8 + jg;
        // hint the next j-block into GL2 while we stream this one
        if (jb + 1 < L_SEQ / 8) {
            __builtin_prefetch(pair + (((long)i0) * L_SEQ + j + 8) * PAIR_D + c4 * 4, 0, 3);
        }
        float4 m1a = make_float4(0.f, 0.f, 0.f, 0.f);
#pragma unroll
        for (int i = 0; i < 16; ++i) {
            const float4 v = *(const float4*)(pair +
                (((long)(i0 + i)) * L_SEQ + j) * PAIR_D + c4 * 4);
            m2acc[i].x += v.x; m2acc[i].y += v.y; m2acc[i].z += v.z; m2acc[i].w += v.w;
            m1a.x += v.x; m1a.y += v.y; m1a.z += v.z; m1a.w += v.w;
        }
        float* dst = m1_sum + (long)j * PAIR_D + c4 * 4;
        atomicAdd(dst + 0, m1a.x); atomicAdd(dst + 1, m1a.y);
        atomicAdd(dst + 2, m1a.z); atomicAdd(dst + 3, m1a.w);
    }

    __shared__ float4 red[8][32];
    const float inv = 1.f / (float)L_SEQ;
    for (int i = 0; i < 16; ++i) {
        red[jg][c4] = m2acc[i];
        __syncthreads();
        if (t < 32) {
            float4 s = red[0][t];
#pragma unroll
            for (int g = 1; g < 8; ++g) {
                s.x += red[g][t].x; s.y += red[g][t].y;
                s.z += red[g][t].z; s.w += red[g][t].w;
            }
            *(float4*)(m2 + (long)(i0 + i) * PAIR_D + t * 4) =
                make_float4(s.x * inv, s.y * inv, s.z * inv, s.w * inv);
        }
        __syncthreads();
    }
}

// ---------------------------------------------------------------------------
// Kernel 2: LayerNorm over seq rows (768 x 384), one block per row
// ---------------------------------------------------------------------------
__global__ void layernorm_kernel(const float* __restrict__ seq,
                                 const float* __restrict__ g,
                                 const float* __restrict__ b,
                                 float* __restrict__ s) {
    const int l = blockIdx.x, t = threadIdx.x;  // 128 threads
    __shared__ float rs[128], rq[128];
    const float* row = seq + (long)l * SEQ_D;
    const float x0 = row[t], x1 = row[t + 128], x2 = row[t + 256];
    rs[t] = x0 + x1 + x2;
    rq[t] = x0 * x0 + x1 * x1 + x2 * x2;
    __syncthreads();
    for (int st = 64; st > 0; st >>= 1) {
        if (t < st) { rs[t] += rs[t + st]; rq[t] += rq[t + st]; }
        __syncthreads();
    }
    const float mu   = rs[0] * (1.f / SEQ_D);
    const float var  = rq[0] * (1.f / SEQ_D) - mu * mu;
    const float rstd = rsqrtf(var + EPSV);
    float* out = s + (long)l * SEQ_D;
    out[t]       = (x0 - mu) * rstd * g[t]       + b[t];
    out[t + 128] = (x1 - mu) * rstd * g[t + 128] + b[t + 128];
    out[t + 256] = (x2 - mu) * rstd * g[t + 256] + b[t + 256];
}

// ---------------------------------------------------------------------------
// Kernel 3: WMMA GEMM  C[M][N] = ascale*(A[M][K] @ W[K][N]) + bias[N]
// Block = 8 waves covering 8 consecutive M-tiles of ONE N-tile. The K x 16
// W panel (<= 24 KB) is staged into LDS with CDNA5 async-to-LDS b128 copies
// (ASYNCcnt), then all 8 waves feed V_WMMA_F32_16X16X4_F32 from LDS.
//   A frag (16x4 f32): lane<16 -> row=lane, K={k,k+1}; lane>=16 -> K={k+2,k+3}
//   B frag (4x16):     same K split, N striped over lanes.
// Grids are exact multiples: M=768 (6 block-rows), N in {256,384}, K in
// {128,256,384} so K*4 transfers divide evenly over 256 threads.
// ---------------------------------------------------------------------------
__global__ void gemm16_kernel(const float* __restrict__ A, const float* __restrict__ W,
                              const float* __restrict__ bias, float* __restrict__ C,
                              int M, int K, int N, float ascale) {
    __shared__ float panel[SEQ_D * 16];   // max K(=384) x 16 = 24 KB
    const int t     = threadIdx.x;        // 256
    const int lane  = t & 31;
    const int wave  = t >> 5;
    const int tilesN = N >> 4;
    const int tn = blockIdx.x % tilesN;
    const int tm = (blockIdx.x / tilesN) * 8 + wave;

    // ---- async-stage W[:, tn*16 : tn*16+16] into LDS (b128 per transfer) ----
    const unsigned lds_base = (unsigned)(uintptr_t)panel;  // low 32b = LDS offset
    const float* wbase = W + tn * 16;
    const int nxfer = (K * 4) >> 8;       // b128 transfers per thread (exact)
    for (int it = 0; it < nxfer; ++it) {
        const int idx = it * 256 + t;     // [0, K*4)
        const int k = idx >> 2, q = idx & 3;
        const float* gp = wbase + (long)k * N + q * 4;
        const unsigned la = lds_base + (unsigned)idx * 16u;
        asm volatile("global_load_async_to_lds_b128 %0, %1, off"
                     :: "v"(la), "v"(gp) : "memory");
    }
    wait_asynccnt0();
    __syncthreads();

    // ---- WMMA main loop ----
    const int r  = lane & 15;
    const int kh = lane >> 4;             // K half-select
    v8f acc = {0.f, 0.f, 0.f, 0.f, 0.f, 0.f, 0.f, 0.f};
    const float* arow = A + (long)(tm * 16 + r) * K;
    const float* brow = panel + r;        // + 16*k
#if defined(__HIP_DEVICE_COMPILE__)
    for (int k = 0; k < K; k += 4) {
        v2f a, b;
        a.x = arow[k + 2 * kh];
        a.y = arow[k + 2 * kh + 1];
        b.x = brow[(k + 2 * kh) * 16];
        b.y = brow[(k + 2 * kh + 1) * 16];
        acc = __builtin_amdgcn_wmma_f32_16x16x4_f32(
            false, a, false, b, (short)0, acc, false, false);
    }
#endif

    // ---- epilogue: uniform A-scale folded here, + bias ----
    const int col = tn * 16 + (lane & 15);
    const float bc = bias[col];
#pragma unroll
    for (int i = 0; i < 8; ++i) {
        const int row = tm * 16 + i + kh * 8;
        C[(long)row * N + col] = acc[i] * ascale + bc;
    }
}

// ---------------------------------------------------------------------------
// Kernel 4: retention core, one block per position l.
//   kv[h,l,k,v] = decay[h]*pi[l,h,k]*pj[l,h,v] + K[l,h,k]*V[l,h,v] -> d_out
//   out_pre[h,l,v] = sum_k Q[l,h,k]*kv ; accumulate per-head sum/sumsq atomics
// ---------------------------------------------------------------------------
__global__ void retention_kernel(const float* __restrict__ Qb, const float* __restrict__ Kb,
                                 const float* __restrict__ Vb, const float* __restrict__ pi,
                                 const float* __restrict__ pj, float* __restrict__ kv_out,
                                 float* __restrict__ out_pre, float* __restrict__ stats) {
    const int l = blockIdx.x;
    const int t = threadIdx.x;   // 256
    const int v = t & 31;
    const int kg = t >> 5;       // 0..7 : group of 4 k's
    __shared__ float part[8][32];
    __shared__ float red[64];

    for (int h = 0; h < NH; ++h) {
        const int base = l * HKD + h * HD;
        const float decay = 1.f - exp2f(-5.f - (float)h);
        const float Vv = Vb[base + v];
        const float Pj = pj[base + v];
        float* kvdst = kv_out + ((long)h * L_SEQ + l) * HD * HD;
        float partial = 0.f;
#pragma unroll
        for (int kk = 0; kk < 4; ++kk) {
            const int k = kg * 4 + kk;
            const float kv = decay * pi[base + k] * Pj + Kb[base + k] * Vv;
            kvdst[k * HD + v] = kv;
            partial += Qb[base + k] * kv;
        }
        part[kg][v] = partial;
        __syncthreads();
        if (t < 32) {
            float o = 0.f;
#pragma unroll
            for (int g = 0; g < 8; ++g) o += part[g][t];
            out_pre[((long)h * L_SEQ + l) * HD + t] = o;
            red[t] = o; red[32 + t] = o * o;
        }
        __syncthreads();
        if (t == 0) {
            float s1 = 0.f, s2 = 0.f;
            for (int i = 0; i < 32; ++i) { s1 += red[i]; s2 += red[32 + i]; }
            atomicAdd(&stats[2 * h], s1);
            atomicAdd(&stats[2 * h + 1], s2);
        }
        __syncthreads();
    }
}

// ---------------------------------------------------------------------------
// Kernel 5: per-head group-norm + gate*sigmoid(gate) -> gated[768][256]
// ---------------------------------------------------------------------------
__global__ void gate_gn_kernel(const float* __restrict__ Gb, const float* __restrict__ out_pre,
                               const float* __restrict__ stats, float* __restrict__ gated) {
    const int l = blockIdx.x;
    const int c = threadIdx.x;  // 256
    const int h = c >> 5;
    const float Ninv = 1.f / (float)(L_SEQ * HD);
    const float mu   = stats[2 * h] * Ninv;
    const float var  = stats[2 * h + 1] * Ninv - mu * mu;
    const float rstd = rsqrtf(var + EPSV);
    const float graw = Gb[(long)l * HKD + c];
    const float gate = 1.f / (1.f + expf(-graw));
    const float sg   = 1.f / (1.f + expf(-gate));
    const float o    = out_pre[((long)h * L_SEQ + l) * HD + (c & 31)];
    gated[(long)l * HKD + c] = gate * sg * (o - mu) * rstd;
}

// ---------------------------------------------------------------------------
extern "C" void kernel_launch(void* const* d_in, const int* in_sizes, int n_in,
                              void* d_out, int out_size, void* d_ws, size_t ws_size,
                              hipStream_t stream) {
    (void)in_sizes; (void)n_in; (void)out_size; (void)ws_size;
    const float* seq  = (const float*)d_in[0];
    const float* pair = (const float*)d_in[1];
    const float* Wq   = (const float*)d_in[2];  const float* bq  = (const float*)d_in[3];
    const float* Wk   = (const float*)d_in[4];  const float* bk  = (const float*)d_in[5];
    const float* Wv   = (const float*)d_in[6];  const float* bv  = (const float*)d_in[7];
    const float* Wbi  = (const float*)d_in[8];  const float* bbi = (const float*)d_in[9];
    const float* Wbj  = (const float*)d_in[10]; const float* bbj = (const float*)d_in[11];
    const float* Wg   = (const float*)d_in[12]; const float* bg  = (const float*)d_in[13];
    const float* Wo   = (const float*)d_in[14]; const float* bo  = (const float*)d_in[15];
    const float* ln_g = (const float*)d_in[16];
    const float* ln_b = (const float*)d_in[17];

    float* out    = (float*)d_out;               // [768][384]
    float* kv_out = out + L_SEQ * SEQ_D;         // [8][768][32][32]

    float* ws   = (float*)d_ws;
    float* m1   = ws; ws += L_SEQ * PAIR_D;      // raw col-sums (atomics)
    float* m2   = ws; ws += L_SEQ * PAIR_D;      // row means
    float* s    = ws; ws += L_SEQ * SEQ_D;       // layernormed seq
    float* Qb   = ws; ws += L_SEQ * HKD;
    float* Kb   = ws; ws += L_SEQ * HKD;
    float* Vb   = ws; ws += L_SEQ * HKD;
    float* Gb   = ws; ws += L_SEQ * HKD;
    float* pi   = ws; ws += L_SEQ * HKD;
    float* pj   = ws; ws += L_SEQ * HKD;
    float* opre = ws; ws += L_SEQ * HKD;
    float* gated= ws; ws += L_SEQ * HKD;
    float* stats= ws; ws += 16;

    zero_buf_kernel<<<(L_SEQ * PAIR_D + 255) / 256, 256, 0, stream>>>(m1, stats);
    pair_reduce_kernel<<<L_SEQ / 16, 256, 0, stream>>>(pair, m1, m2);
    layernorm_kernel<<<L_SEQ, 128, 0, stream>>>(seq, ln_g, ln_b, s);

    auto launch_gemm = [&](const float* A, const float* W, const float* B, float* C,
                           int M, int K, int N, float asc) {
        const int blocks = (M / 128) * (N / 16);   // 8 M-tiles per block
        gemm16_kernel<<<blocks, 256, 0, stream>>>(A, W, B, C, M, K, N, asc);
    };
    launch_gemm(s,  Wq,  bq,  Qb, L_SEQ, SEQ_D,  HKD, 1.f);
    launch_gemm(s,  Wk,  bk,  Kb, L_SEQ, SEQ_D,  HKD, 1.f);
    launch_gemm(s,  Wv,  bv,  Vb, L_SEQ, SEQ_D,  HKD, 1.f);
    launch_gemm(s,  Wg,  bg,  Gb, L_SEQ, SEQ_D,  HKD, 1.f);
    launch_gemm(m1, Wbi, bbi, pi, L_SEQ, PAIR_D, HKD, 1.f / (float)L_SEQ);
    launch_gemm(m2, Wbj, bbj, pj, L_SEQ, PAIR_D, HKD, 1.f);

    retention_kernel<<<L_SEQ, 256, 0, stream>>>(Qb, Kb, Vb, pi, pj, kv_out, opre, stats);
    gate_gn_kernel<<<L_SEQ, 256, 0, stream>>>(Gb, opre, stats, gated);
    launch_gemm(gated, Wo, bo, out, L_SEQ, HKD, SEQ_D, 1.f);
}